// SNN_18133351924315
// MI455X (gfx1250) — compile-verified
//
#include <hip/hip_runtime.h>
#include <math.h>

typedef __attribute__((ext_vector_type(2))) float v2f;
typedef __attribute__((ext_vector_type(8))) float v8f;

#define B_   16
#define N_   2048
#define NFF_ 512
#define D_   4
#define BN_  (B_ * N_)

// ---------------------------------------------------------------------------
// Kernel 0: zero the 32 per-batch scalar accumulators (s_p[16], s_d[16])
// ---------------------------------------------------------------------------
__global__ void snn_zero(float* __restrict__ s) {
    s[threadIdx.x] = 0.0f;   // launched with 32 threads
}

// ---------------------------------------------------------------------------
// Kernel 1: elementwise prep over [B,N]
//   X   = sigmoid((V - (1 + th))/0.2)              -> d_out slot 0
//   a   = sum_d Xd*(1 + stp*0.95 - 0.3*Xd)         -> ws Aeff
//   s_p[b] += sum_i a*xbar_pre_int ; s_d[b] += sum_i a*Xd0
// Each 256-thread block covers a contiguous chunk inside one batch row
// (256 | 2048), so one atomicAdd pair per block.
// ---------------------------------------------------------------------------
__global__ void snn_prep(const float* __restrict__ V,
                         const float* __restrict__ th,
                         const float* __restrict__ Xbuf,      // [D,B,N]
                         const float* __restrict__ stp,       // [D,B,N]
                         const float* __restrict__ xpre_int,  // [B,N]
                         float* __restrict__ Xout,            // [B,N]
                         float* __restrict__ Aeff,            // [B,N]
                         float* __restrict__ s_p,
                         float* __restrict__ s_d) {
    int idx = blockIdx.x * 256 + threadIdx.x;            // < B*N
    float v = V[idx];
    float t = th[idx];
    float x = 1.0f / (1.0f + __expf(-(v - (1.0f + t)) * 5.0f));  // 1/SURR_BETA = 5
    Xout[idx] = x;

    float a = 0.0f, xd0 = 0.0f;
#pragma unroll
    for (int d = 0; d < D_; ++d) {
        float xd = Xbuf[d * BN_ + idx];
        float st = stp[d * BN_ + idx];
        a += xd * (1.0f + st * 0.95f - 0.3f * xd);
        if (d == 0) xd0 = xd;
    }
    Aeff[idx] = a;

    __shared__ float r1[256], r2[256];
    r1[threadIdx.x] = a * xpre_int[idx];
    r2[threadIdx.x] = a * xd0;
    __syncthreads();
#pragma unroll
    for (int s = 128; s > 0; s >>= 1) {
        if ((int)threadIdx.x < s) {
            r1[threadIdx.x] += r1[threadIdx.x + s];
            r2[threadIdx.x] += r2[threadIdx.x + s];
        }
        __syncthreads();
    }
    if (threadIdx.x == 0) {
        int b = idx / N_;
        atomicAdd(&s_p[b], r1[0]);
        atomicAdd(&s_d[b], r2[0]);
    }
}

// ---------------------------------------------------------------------------
// Kernel 2: FF scalar reductions: s_p[b] += FF*xpre_ff, s_d[b] += FF*FF
// ---------------------------------------------------------------------------
__global__ void snn_prep_ff(const float* __restrict__ FF,
                            const float* __restrict__ xpre_ff,
                            float* __restrict__ s_p,
                            float* __restrict__ s_d) {
    int idx = blockIdx.x * 256 + threadIdx.x;            // < B*NFF
    float f = FF[idx];
    __shared__ float r1[256], r2[256];
    r1[threadIdx.x] = f * xpre_ff[idx];
    r2[threadIdx.x] = f * f;
    __syncthreads();
#pragma unroll
    for (int s = 128; s > 0; s >>= 1) {
        if ((int)threadIdx.x < s) {
            r1[threadIdx.x] += r1[threadIdx.x + s];
            r2[threadIdx.x] += r2[threadIdx.x + s];
        }
        __syncthreads();
    }
    if (threadIdx.x == 0) {
        int b = idx / NFF_;
        atomicAdd(&s_p[b], r1[0]);
        atomicAdd(&s_d[b], r2[0]);
    }
}

// ---------------------------------------------------------------------------
// Kernel 3: WMMA GEMM + fused epilogue.
//   current[16,N] = Aeff[16,2048] @ W_int[2048,N] + FF[16,512] @ W_ff[512,N]
//   V_new = 0.9*V*(1-X) + current + 0.01*X*s_p[b] - 0.012*xbar_post*s_d[b]
// One block (256 thr = 8 waves) per 16-wide n-tile; wave w takes K-chunk w
// (256 of W_int's K, 64 of W_ff's K), partial 16x16 tiles reduced in LDS.
//
// V_WMMA_F32_16X16X4_F32 operand layout (wave32):
//   A 16x4 : lane L holds row m=L%16; VGPR0/1 = K={2*(L/16), 2*(L/16)+1}
//   B 4x16 : lane L holds col n=L%16; VGPR0/1 = K rows {2*(L/16), 2*(L/16)+1}
//   C 16x16: lane L col n=L%16; VGPR v -> row m = v + 8*(L/16)
// ---------------------------------------------------------------------------
__global__ void snn_gemm(const float* __restrict__ Aeff,      // [16,2048]
                         const float* __restrict__ FF,        // [16,512]
                         const float* __restrict__ W_int,     // [2048,2048]
                         const float* __restrict__ W_ff,      // [512,2048]
                         const float* __restrict__ V,         // [16,2048]
                         const float* __restrict__ xbar_post, // [16,2048]
                         const float* __restrict__ s_p,       // [16]
                         const float* __restrict__ s_d,       // [16]
                         const float* __restrict__ Xout,      // [16,2048]
                         float* __restrict__ Vout) {          // [16,2048]
    const int tile_n = blockIdx.x;            // 0..127
    const int wave   = threadIdx.x >> 5;      // 0..7
    const int lane   = threadIdx.x & 31;
    const int lh     = lane & 15;             // m for A rows / n for B cols
    const int hi     = lane >> 4;             // K-half selector
    const int j      = tile_n * 16 + lh;      // global column

    v8f acc = {};

    // ---- recurrent part: K-chunk [wave*256, wave*256+256) of 2048 ----
    {
        const int k0 = wave * 256;
        const float* arow = Aeff + lh * N_;
#pragma unroll 4
        for (int k = k0; k < k0 + 256; k += 4) {
            v2f a = *(const v2f*)(arow + k + 2 * hi);
            v2f b;
            b.x = W_int[(k + 2 * hi)     * N_ + j];
            b.y = W_int[(k + 2 * hi + 1) * N_ + j];
            acc = __builtin_amdgcn_wmma_f32_16x16x4_f32(
                false, a, false, b, (short)0, acc, false, false);
        }
    }
    // ---- feedforward part: K-chunk [wave*64, wave*64+64) of 512 ----
    {
        const int k0 = wave * 64;
        const float* frow = FF + lh * NFF_;
#pragma unroll 4
        for (int k = k0; k < k0 + 64; k += 4) {
            v2f a = *(const v2f*)(frow + k + 2 * hi);
            v2f b;
            b.x = W_ff[(k + 2 * hi)     * N_ + j];
            b.y = W_ff[(k + 2 * hi + 1) * N_ + j];
            acc = __builtin_amdgcn_wmma_f32_16x16x4_f32(
                false, a, false, b, (short)0, acc, false, false);
        }
    }

    // ---- reduce the 8 partial tiles through LDS ----
    __shared__ float red[8 * 256];
#pragma unroll
    for (int v = 0; v < 8; ++v)
        red[wave * 256 + v * 32 + lane] = acc[v];
    __syncthreads();

    const int t = threadIdx.x;                // one thread per tile element
    float sum = 0.0f;
#pragma unroll
    for (int w = 0; w < 8; ++w) sum += red[w * 256 + t];

    // element coords from LDS layout: v = t/32, lane' = t%32
    const int v  = t >> 5;
    const int ln = t & 31;
    const int m  = v + 8 * (ln >> 4);         // batch index b
    const int nn = ln & 15;
    const int idx = m * N_ + tile_n * 16 + nn;

    const float x  = Xout[idx];
    const float vn = 0.9f * V[idx] * (1.0f - x) + sum
                   + 0.01f  * x * s_p[m]
                   - 0.012f * xbar_post[idx] * s_d[m];
    Vout[idx] = vn;
}

// ---------------------------------------------------------------------------
extern "C" void kernel_launch(void* const* d_in, const int* in_sizes, int n_in,
                              void* d_out, int out_size, void* d_ws, size_t ws_size,
                              hipStream_t stream) {
    // setup_inputs order:
    // 0:FF 1:V 2:th_adapt 3:Xbuf 4:stp_state 5:xbar_pre_int 6:xbar_post
    // 7:xbar_pre_ff 8:W_int 9:W_ff
    const float* FF        = (const float*)d_in[0];
    const float* V         = (const float*)d_in[1];
    const float* th        = (const float*)d_in[2];
    const float* Xbuf      = (const float*)d_in[3];
    const float* stp       = (const float*)d_in[4];
    const float* xpre_int  = (const float*)d_in[5];
    const float* xbar_post = (const float*)d_in[6];
    const float* xpre_ff   = (const float*)d_in[7];
    const float* W_int     = (const float*)d_in[8];
    const float* W_ff      = (const float*)d_in[9];

    float* out  = (float*)d_out;
    float* Xout = out;              // [B,N]  (output slot 0)
    float* Vout = out + BN_;        // [B,N]  (output slot 1)

    float* ws   = (float*)d_ws;
    float* Aeff = ws;               // B*N floats
    float* s_p  = ws + BN_;         // 16 floats
    float* s_d  = ws + BN_ + 16;    // 16 floats

    snn_zero<<<1, 32, 0, stream>>>(s_p);  // zeros s_p[16] + s_d[16] (contiguous)

    snn_prep<<<BN_ / 256, 256, 0, stream>>>(V, th, Xbuf, stp, xpre_int,
                                            Xout, Aeff, s_p, s_d);

    snn_prep_ff<<<(B_ * NFF_) / 256, 256, 0, stream>>>(FF, xpre_ff, s_p, s_d);

    snn_gemm<<<N_ / 16, 256, 0, stream>>>(Aeff, FF, W_int, W_ff, V,
                                          xbar_post, s_p, s_d, Xout, Vout);
}